// LDLLoss_86431921864957
// MI455X (gfx1250) — compile-verified
//
#include <hip/hip_runtime.h>
#include <hip/hip_bf16.h>
#include <math.h>

// ---------------------------------------------------------------------------
// LDL loss, shapes fixed by reference: N=16, C=3, H=W=512, ksize=7.
// Memory-bound: ~134MB total traffic -> ~5.8us at 23.3 TB/s HBM.
// residual (16.8MB) parked in ws, re-read from 192MB L2 in the stencil pass.
// ---------------------------------------------------------------------------

#define N_IMG   16
#define HW      (512 * 512)          // 262144 per channel plane
#define HWC4    (HW / 4)
#define TS      64                   // loss-pass tile
#define HALO    3
#define LW      (TS + 2 * HALO)      // 70
#define LPITCH  72                   // LDS row pitch (pad to dodge bank conflicts)

typedef __attribute__((ext_vector_type(2))) float v2f;
typedef __attribute__((ext_vector_type(8))) float v8f;

typedef __attribute__((address_space(1))) int gint_t;
typedef __attribute__((address_space(3))) int lint_t;

// ----- CDNA5 async global->LDS copy (ASYNCcnt path) ------------------------
#if __has_builtin(__builtin_amdgcn_global_load_async_to_lds_b32)
#define HAVE_ASYNC_LDS 1
__device__ __forceinline__ void async_copy_b32(const float* g, float* l) {
  __builtin_amdgcn_global_load_async_to_lds_b32(
      (gint_t*)(g),
      (lint_t*)(l),
      0, 0);
}
__device__ __forceinline__ void wait_async0() {
#if __has_builtin(__builtin_amdgcn_s_wait_asynccnt)
  __builtin_amdgcn_s_wait_asynccnt(0);
#else
  asm volatile("s_wait_asynccnt 0" ::: "memory");
#endif
}
#else
#define HAVE_ASYNC_LDS 0
#endif

// ---------------------------------------------------------------------------
__global__ void k_zero(float* ws) { ws[threadIdx.x] = 0.0f; }

// residual = sum_c |t-p|; also per-image sum(r), sum(r^2) partials
__global__ void k_residual(const float* __restrict__ pred,
                           const float* __restrict__ targ,
                           float* __restrict__ res,
                           float* __restrict__ sumr,
                           float* __restrict__ sumr2) {
  const int n   = blockIdx.y;
  const int tid = threadIdx.x;
  const int i4  = blockIdx.x * 256 + tid;          // float4 index in plane
  const size_t base = (size_t)n * 3 * HW;

  const float4* P = (const float4*)(pred + base);
  const float4* T = (const float4*)(targ + base);

  float4 a = P[i4];
  float4 b = T[i4];
  float4 r;
  r.x = fabsf(b.x - a.x); r.y = fabsf(b.y - a.y);
  r.z = fabsf(b.z - a.z); r.w = fabsf(b.w - a.w);
  a = P[i4 + HWC4];       b = T[i4 + HWC4];
  r.x += fabsf(b.x - a.x); r.y += fabsf(b.y - a.y);
  r.z += fabsf(b.z - a.z); r.w += fabsf(b.w - a.w);
  a = P[i4 + 2 * HWC4];   b = T[i4 + 2 * HWC4];
  r.x += fabsf(b.x - a.x); r.y += fabsf(b.y - a.y);
  r.z += fabsf(b.z - a.z); r.w += fabsf(b.w - a.w);

  ((float4*)(res + (size_t)n * HW))[i4] = r;

  float s  = r.x + r.y + r.z + r.w;
  float s2 = r.x * r.x + r.y * r.y + r.z * r.z + r.w * r.w;
  for (int m = 16; m; m >>= 1) {
    s  += __shfl_xor(s,  m, 32);
    s2 += __shfl_xor(s2, m, 32);
  }
  __shared__ float ls[8], ls2[8];
  const int wid = tid >> 5;
  if ((tid & 31) == 0) { ls[wid] = s; ls2[wid] = s2; }
  __syncthreads();
  if (tid == 0) {
    float ta = 0.0f, tb = 0.0f;
    for (int i = 0; i < 8; ++i) { ta += ls[i]; tb += ls2[i]; }
    atomicAdd(&sumr[n], ta);
    atomicAdd(&sumr2[n], tb);
  }
}

// patch_w[n] = var(residual over HW, ddof=1) ^ 0.2
__global__ void k_patchw(const float* __restrict__ sumr,
                         const float* __restrict__ sumr2,
                         float* __restrict__ patchw) {
  const int t = threadIdx.x;
  if (t < N_IMG) {
    const float M = (float)HW;
    float var = (sumr2[t] - sumr[t] * sumr[t] / M) / (M - 1.0f);
    patchw[t] = powf(fmaxf(var, 0.0f), 0.2f);
  }
}

// 7x7 local unbiased variance (separable box sums) * residual * patch_w,
// reduced to a single accumulator.
__global__ void __launch_bounds__(256) k_loss(const float* __restrict__ res,
                                              const float* __restrict__ patchw,
                                              float* __restrict__ acc) {
  __shared__ float tr[LW * LPITCH];    // residual tile + halo (70x70 in 70x72)
  __shared__ float hs[LW * TS];        // horizontal 7-sums of r   (70x64)
  __shared__ float hq[LW * TS];        // horizontal 7-sums of r^2 (70x64)
  __shared__ float red[8];

  const int tid = threadIdx.x;
  const int n   = blockIdx.z;
  const int tx0 = blockIdx.x * TS;
  const int ty0 = blockIdx.y * TS;
  const float* gimg = res + (size_t)n * HW;

  // ---- Phase A: reflect-padded tile load (async global -> LDS) ----
  for (int i = tid; i < LW * LW; i += 256) {
    const int ly = i / LW, lx = i - ly * LW;
    int gy = ty0 + ly - HALO;
    int gx = tx0 + lx - HALO;
    gy = (gy < 0) ? -gy : ((gy > 511) ? 1022 - gy : gy);
    gx = (gx < 0) ? -gx : ((gx > 511) ? 1022 - gx : gx);
#if HAVE_ASYNC_LDS
    async_copy_b32(gimg + (gy * 512 + gx), &tr[ly * LPITCH + lx]);
#else
    tr[ly * LPITCH + lx] = gimg[gy * 512 + gx];
#endif
  }
#if HAVE_ASYNC_LDS
  wait_async0();
#endif
  __syncthreads();

  // ---- Phase B: horizontal 7-wide sums of r and r^2 ----
  for (int i = tid; i < LW * TS; i += 256) {
    const int y = i / TS, x = i - y * TS;
    const float* row = &tr[y * LPITCH + x];
    float s = 0.0f, q = 0.0f;
    #pragma unroll
    for (int dx = 0; dx < 7; ++dx) {
      const float v = row[dx];
      s += v;
      q += v * v;
    }
    hs[i] = s;
    hq[i] = q;
  }
  __syncthreads();

  // ---- Phase C: vertical 7-sums -> pixel_w -> weighted contribution ----
  float local = 0.0f;
  for (int i = tid; i < TS * TS; i += 256) {
    const int y = i / TS, x = i - y * TS;
    float s = 0.0f, q = 0.0f;
    #pragma unroll
    for (int dy = 0; dy < 7; ++dy) {
      s += hs[(y + dy) * TS + x];
      q += hq[(y + dy) * TS + x];
    }
    const float pixw = (q - s * s * (1.0f / 49.0f)) * (1.0f / 48.0f);
    const float rc   = tr[(y + HALO) * LPITCH + (x + HALO)];
    local += pixw * rc;
  }
  local *= patchw[n];

  // ---- Phase D: wave reduction via V_WMMA_F32_16X16X4_F32 ones-matmul ----
  // A[m][0]=p(lane m), A[m][2]=p(lane m+16), A[m][1]=A[m][3]=0; B=ones.
  // D[m][*] = p(m)+p(m+16); sum of 8 D VGPRs gives half-sums in lane groups;
  // one xor-16 shuffle completes the 32-lane sum (EXEC is all-1s here).
  float wsum;
#if __has_builtin(__builtin_amdgcn_wmma_f32_16x16x4_f32)
  {
    v2f a; a[0] = local; a[1] = 0.0f;
    v2f b; b[0] = 1.0f;  b[1] = 1.0f;
    v8f c = {};
    c = __builtin_amdgcn_wmma_f32_16x16x4_f32(false, a, false, b,
                                              (short)0, c, false, false);
    float t = c[0] + c[1] + c[2] + c[3] + c[4] + c[5] + c[6] + c[7];
    t += __shfl_xor(t, 16, 32);
    wsum = t;
  }
#else
  {
    float t = local;
    for (int m = 16; m; m >>= 1) t += __shfl_xor(t, m, 32);
    wsum = t;
  }
#endif
  if ((tid & 31) == 0) red[tid >> 5] = wsum;
  __syncthreads();
  if (tid == 0) {
    float t = 0.0f;
    for (int i = 0; i < 8; ++i) t += red[i];
    atomicAdd(acc, t);
  }
}

__global__ void k_final(const float* __restrict__ acc, float* __restrict__ out) {
  out[0] = acc[0] * (1.0f / (float)(N_IMG * 3 * HW));  // LOSS_WEIGHT = 1
}

// ---------------------------------------------------------------------------
extern "C" void kernel_launch(void* const* d_in, const int* in_sizes, int n_in,
                              void* d_out, int out_size, void* d_ws, size_t ws_size,
                              hipStream_t stream) {
  const float* pred = (const float*)d_in[0];
  const float* targ = (const float*)d_in[1];
  float* out = (float*)d_out;
  float* ws  = (float*)d_ws;

  float* sumr   = ws;        // [0,16)
  float* sumr2  = ws + 16;   // [16,32)
  float* patchw = ws + 32;   // [32,48)
  float* acc    = ws + 48;   // [48]
  float* res    = ws + 64;   // residual: 16 * 262144 floats (16.8 MB)

  k_zero<<<1, 64, 0, stream>>>(ws);
  k_residual<<<dim3(HWC4 / 256, N_IMG), 256, 0, stream>>>(pred, targ, res, sumr, sumr2);
  k_patchw<<<1, 32, 0, stream>>>(sumr, sumr2, patchw);
  k_loss<<<dim3(512 / TS, 512 / TS, N_IMG), 256, 0, stream>>>(res, patchw, acc);
  k_final<<<1, 1, 0, stream>>>(acc, out);
}